// kernel_generated_4_21887153341312
// MI455X (gfx1250) — compile-verified
//
#include <hip/hip_runtime.h>

// Fused "im2col + GEMM" for the rolled/unfolded conv reference.
//   y[i,n,o] = sum_{j,k,l} w[i,j,k,l] * T[(j,k,l),(n,o)]
// M=256, N=196 (14x14 pixels), K=1152 (3*3*128), all fp32.
// ~1.6MB traffic / ~116 MFLOP -> latency bound on MI455X. Single fused
// kernel: per-block LDS-staged B tiles, fp32 WMMA (16x16x4) for
// reference-matching precision, 4 independent accumulators to break the
// 288-deep WMMA D->C dependency chain.

typedef __attribute__((ext_vector_type(2))) float v2f;
typedef __attribute__((ext_vector_type(8))) float v8f;

#define KTOT        1152
#define KCHUNK      576
#define LDS_STRIDE  580    // 576 + 4 pad -> bank spread for col-major tile
#define NPIX        196
#define NTILES      13

static __device__ __forceinline__ v8f wmma_f32(v2f a, v2f b, v8f c) {
    return __builtin_amdgcn_wmma_f32_16x16x4_f32(
        /*neg_a=*/false, a, /*neg_b=*/false, b,
        /*c_mod=*/(short)0, c, /*reuse_a=*/false, /*reuse_b=*/false);
}

__global__ __launch_bounds__(512) void fused_roll_conv_wmma(
    const float* __restrict__ x,   // (256,14,14) flat: c*196 + h*14 + w
    const float* __restrict__ w,   // (256,1152)  flat: i*1152 + (j*3+k)*128 + l
    float* __restrict__ y)         // (256,196)   flat: i*196 + n*14 + o
{
    __shared__ float sT[16 * LDS_STRIDE];   // B chunk: [col 0..15][kk 0..575]

    const int tid   = threadIdx.x;
    const int lane  = tid & 31;
    const int wv    = tid >> 5;          // wave id 0..15 == M-tile == fill col
    const int ntile = blockIdx.x;        // 0..12

    const int colN = lane & 15;          // A: M row / B,C,D: N col
    const int koff = (lane >> 4) << 1;   // K sub-offset 0 or 2 per half-wave

    const int   arow  = (wv << 4) + colN;                     // i row of A
    const float* aptr  = w + (size_t)arow * KTOT + koff;
    const float* bbase = &sT[colN * LDS_STRIDE + koff];

    // ---- per-wave (scalar) pixel-position constants for the fill phase ----
    const int fp = ntile * 16 + wv;      // this wave fills column `wv`
    const int fn = fp / 14;
    const int fo = fp - fn * 14;
    const int fb1 = (fo + 13) % 14;      // (o-1) mod 14
    const bool fvalid = (fp < NPIX);
    float* sTcol = &sT[wv * LDS_STRIDE];

    v8f acc0 = {0.f,0.f,0.f,0.f,0.f,0.f,0.f,0.f};
    v8f acc1 = acc0, acc2 = acc0, acc3 = acc0;

    for (int kc = 0; kc < KTOT; kc += KCHUNK) {
        // pull this chunk's A-fragment stream toward the caches while we fill
        __builtin_prefetch(aptr + kc, 0, 3);

        // ---- cooperative im2col fill: wave wv fills column wv,
        //      lane-consecutive kk -> conflict-free LDS stores (18 uniform trips)
        for (int q = 0; q < KCHUNK / 32; ++q) {
            int kk    = lane + (q << 5);
            int kflat = kc + kk;
            int l  = kflat & 127;
            int jk = kflat >> 7;
            int k  = jk % 3;
            int j  = jk / 3;
            float v = 0.f;
            if (fvalid) {
                int a = fn + j - 1;                      // second unfold (H)
                if (0 <= a && a < 14) {
                    int s0 = fo + k;                     // first unfold (W), term 0
                    if (s0 >= 1 && s0 <= 14) {
                        int w0 = (s0 + 12) % 14;         // (o+k-2) mod 14
                        v += x[l * NPIX + a * 14 + w0];
                    }
                    int s1 = fb1 + k;                    // term 1: rolled half
                    if (s1 >= 1 && s1 <= 14) {
                        int w1 = (s1 + 12) % 14;         // (b1+k-2) mod 14
                        v += x[(128 + l) * NPIX + a * 14 + w1];
                    }
                }
            }
            sTcol[kk] = v;
        }
        __syncthreads();

        // ---- 144 fp32 WMMAs over the chunk, 4 independent accumulators ----
        const float* ap = aptr + kc;
        #pragma unroll 2
        for (int kk = 0; kk < KCHUNK; kk += 16) {
            v2f a0 = *(const v2f*)(ap + kk);
            v2f b0 = *(const v2f*)(bbase + kk);
            v2f a1 = *(const v2f*)(ap + kk + 4);
            v2f b1 = *(const v2f*)(bbase + kk + 4);
            v2f a2 = *(const v2f*)(ap + kk + 8);
            v2f b2 = *(const v2f*)(bbase + kk + 8);
            v2f a3 = *(const v2f*)(ap + kk + 12);
            v2f b3 = *(const v2f*)(bbase + kk + 12);
            acc0 = wmma_f32(a0, b0, acc0);
            acc1 = wmma_f32(a1, b1, acc1);
            acc2 = wmma_f32(a2, b2, acc2);
            acc3 = wmma_f32(a3, b3, acc3);
        }
        __syncthreads();
    }

    // ---- merge accumulators, store C/D tile:
    //      lane holds col N=lane%16, rows r + 8*(lane/16) across 8 VGPRs
    v8f acc = (acc0 + acc1) + (acc2 + acc3);
    int p = ntile * 16 + colN;
    if (p < NPIX) {
        int half = (lane >> 4) << 3;    // 0 or 8
        #pragma unroll
        for (int r = 0; r < 8; ++r) {
            int i = (wv << 4) + half + r;
            y[(size_t)i * NPIX + p] = acc[r];
        }
    }
}

extern "C" void kernel_launch(void* const* d_in, const int* in_sizes, int n_in,
                              void* d_out, int out_size, void* d_ws, size_t ws_size,
                              hipStream_t stream) {
    const float* x = (const float*)d_in[0];   // (1,256,14,14) fp32
    const float* w = (const float*)d_in[1];   // (256,3,3,128) fp32
    float* y = (float*)d_out;                 // (1,256,14,14) fp32
    (void)in_sizes; (void)n_in; (void)out_size; (void)d_ws; (void)ws_size;
    fused_roll_conv_wmma<<<dim3(NTILES), dim3(512), 0, stream>>>(x, w, y);
}